// System_25984552140968
// MI455X (gfx1250) — compile-verified
//
#include <hip/hip_runtime.h>
#include <math.h>

// ---------------------------------------------------------------------------
// Persistent cooperative SDE kernel for MI455X (gfx1250, wave32).
//  * 1 path per thread, 256 blocks x 256 threads = 65536 paths, state in VGPRs
//  * per-step MLP input layer via V_WMMA_F32_16X16X4_F32 (two 16-path tiles/wave)
//  * BatchNorm over the full batch: LDS atomics -> global atomics -> counter
//    based grid barrier (s_sleep spin, agent-scope acquire/release)
//  * second layer via wave-private LDS tile [hidden][path] (4x ds_store_b128
//    scatter + 16x conflict-free ds_load_b32 gather) instead of a 64-shuffle
//    butterfly: ~20 DS ops/step instead of ~73
//  * dW streamed as aligned float4 (B128) every 4 steps + global_prefetch;
//    whole dW (131 MB) fits the 192 MB L2, steady state is L2-resident
// Reference-constant folds (exact): b1 cancels under BN; NU=0 => X==1,dX==0;
// sigma_t==0 => drift_Z = Delta; QEXP=2 => sign(Z)(|Z|/Lam)^2 = Z|Z|/Lam^2;
// C_Z=1, C_LAM=0.09, ALPHA=1.
// ---------------------------------------------------------------------------

typedef __attribute__((ext_vector_type(2))) float v2f;
typedef __attribute__((ext_vector_type(8))) float v8f;

#define NPATH 65536
#define TSN   500
#define HIDN  15
#define NINN  4
#define BLOCK 256
#define NBLK  256
#define WPB   8          // waves per block
#define YSTRIDE 36       // [hidden][path] row stride (floats): 144B, 16B-aligned

static constexpr float kDT = 0.002f;

__device__ __forceinline__ float shfl32(float v, int src) { return __shfl(v, src, 32); }
__device__ __forceinline__ float sxor32(float v, int m)   { return __shfl_xor(v, m, 32); }

__global__ __launch_bounds__(BLOCK)
void sde_persistent(const float* __restrict__ dW,
                    const float* __restrict__ W1,
                    const float* __restrict__ g1,
                    const float* __restrict__ be1,
                    const float* __restrict__ W2,
                    const float* __restrict__ b2,
                    float* __restrict__ out,
                    float* __restrict__ gsum,
                    float* __restrict__ gsq,
                    unsigned int* __restrict__ done)
{
    __shared__ float bs_sum[16];
    __shared__ float bs_sq[16];
    __shared__ float ybuf[WPB][16 * YSTRIDE];   // wave-private [hidden][path] tiles

    const int tid   = threadIdx.x;
    const int lane  = tid & 31;
    const int wave  = tid >> 5;
    const int path  = blockIdx.x * BLOCK + tid;
    const int n     = lane & 15;              // hidden unit this lane owns (WMMA C layout)
    const bool loH  = lane < 16;
    const bool nv   = n < HIDN;               // column 15 is zero padding
    const int p0    = loH ? 0 : 8;            // first row this lane's C VGPRs cover
    const unsigned int nblk = gridDim.x;

    // per-path state in registers (X is provably constant 1.0: NU=0, X_RATE=0)
    float Wst = 0.f, Xst = 1.f, XIst = 0.f, Dst = 0.f, Zst = 0.f;

    float dwbuf[4];
    const float* dWrow = dW + (size_t)path * TSN;

    for (int j = 0; j <= TSN; ++j) {
        if (tid < 16) { bs_sum[tid] = 0.f; bs_sq[tid] = 0.f; }
        __syncthreads();

        // ---- B matrix = W1[j]^T, 4x16 f32 (lanes 0-15: K=0,1 ; lanes 16-31: K=2,3)
        const float* W1j = W1 + (size_t)j * (HIDN * NINN);
        const int kh = loH ? 0 : 2;
        v2f B;
        B.x = nv ? W1j[n * NINN + kh]     : 0.f;
        B.y = nv ? W1j[n * NINN + kh + 1] : 0.f;

        // ---- two 16-path tiles: A = [W X ; XI Delta] gathered from owner lanes
        v8f c0 = {}, c1 = {};
        {
            const int s0 = n;                 // paths base+0..15
            float sw = shfl32(Wst,  s0), sx = shfl32(Xst, s0);
            float si = shfl32(XIst, s0), sd = shfl32(Dst, s0);
            v2f A; A.x = loH ? sw : si; A.y = loH ? sx : sd;
            c0 = __builtin_amdgcn_wmma_f32_16x16x4_f32(false, A, false, B,
                                                       (short)0, c0, false, false);
        }
        {
            const int s1 = n + 16;            // paths base+16..31
            float sw = shfl32(Wst,  s1), sx = shfl32(Xst, s1);
            float si = shfl32(XIst, s1), sd = shfl32(Dst, s1);
            v2f A; A.x = loH ? sw : si; A.y = loH ? sx : sd;
            c1 = __builtin_amdgcn_wmma_f32_16x16x4_f32(false, A, false, B,
                                                       (short)0, c1, false, false);
        }

        // ---- BN partial sums: lane owns one hidden column over 8 rows per tile
        float s = 0.f, q = 0.f;
#pragma unroll
        for (int r = 0; r < 8; ++r) {
            float a = c0[r], bvl = c1[r];
            s += a + bvl; q += a * a + bvl * bvl;
        }
        s += sxor32(s, 16);
        q += sxor32(q, 16);
        if (loH && nv) {
            atomicAdd(&bs_sum[n], s);
            atomicAdd(&bs_sq[n],  q);
        }
        __syncthreads();

        // ---- block -> global accumulators, then grid barrier on done[j]
        if (wave == 0 && tid < HIDN) {
            atomicAdd(&gsum[j * 16 + tid], bs_sum[tid]);
            atomicAdd(&gsq [j * 16 + tid], bs_sq [tid]);
        }
        __threadfence();
        __syncthreads();
        if (tid == 0) {
            __hip_atomic_fetch_add(&done[j], 1u, __ATOMIC_RELEASE,
                                   __HIP_MEMORY_SCOPE_AGENT);
            while (__hip_atomic_load(&done[j], __ATOMIC_ACQUIRE,
                                     __HIP_MEMORY_SCOPE_AGENT) < nblk)
                __builtin_amdgcn_s_sleep(1);
        }
        __syncthreads();

        // ---- finalize BN (b1 cancels exactly), ReLU, output-layer weights
        float scale = 0.f, shift = 0.f, w2n = 0.f;
        if (nv) {
            float gs = __hip_atomic_load(&gsum[j * 16 + n], __ATOMIC_RELAXED,
                                         __HIP_MEMORY_SCOPE_AGENT);
            float gq = __hip_atomic_load(&gsq [j * 16 + n], __ATOMIC_RELAXED,
                                         __HIP_MEMORY_SCOPE_AGENT);
            float mu  = gs * (1.0f / NPATH);
            float var = gq * (1.0f / NPATH) - mu * mu;
            float rs  = rsqrtf(var + 1e-5f);
            scale = rs * g1[(size_t)j * HIDN + n];
            shift = be1[(size_t)j * HIDN + n] - mu * scale;
            w2n   = W2[(size_t)j * HIDN + n];
        }

        float y0[8], y1[8];
#pragma unroll
        for (int r = 0; r < 8; ++r) {
            y0[r] = fmaxf(c0[r] * scale + shift, 0.f) * w2n;
            y1[r] = fmaxf(c1[r] * scale + shift, 0.f) * w2n;
        }

        // ---- scatter per-lane column products into wave-private LDS tile:
        //      layout [hidden n][path-in-wave p], lane writes 2x 8-path runs
        //      per tile as ds_store_b128 pairs (addresses 16B-aligned)
        {
            float* yb = &ybuf[wave][n * YSTRIDE];
            *(float4*)(yb + p0)      = make_float4(y0[0], y0[1], y0[2], y0[3]);
            *(float4*)(yb + p0 + 4)  = make_float4(y0[4], y0[5], y0[6], y0[7]);
            *(float4*)(yb + p0 + 16) = make_float4(y1[0], y1[1], y1[2], y1[3]);
            *(float4*)(yb + p0 + 20) = make_float4(y1[4], y1[5], y1[6], y1[7]);
        }
        // gather own path's 16 hidden contributions (stride-36: conflict-free)
        float zacc = 0.f;
#pragma unroll
        for (int h = 0; h < 16; ++h) zacc += ybuf[wave][h * YSTRIDE + lane];
        const float sZ = zacc + b2[j];

        if (j == 0) {
            Zst = sZ;                         // Z_0 = mlp(params[0], in_0)
        } else {
            const int idx = j - 1;
            if ((idx & 3) == 0) {             // aligned B128 every 4 steps
                const float4 v = *(const float4*)(dWrow + idx);
                dwbuf[0] = v.x; dwbuf[1] = v.y; dwbuf[2] = v.z; dwbuf[3] = v.w;
                __builtin_prefetch(dWrow + idx + 4, 0, 1);   // global_prefetch_b8
            }
            const float dw = dwbuf[idx & 3];
            const float x2    = Xst * Xst;            // == 1
            const float sigxi = sqrtf(x2);
            const float dZ  = Dst * kDT + sZ * dw;    // C_Z=1, sigma_t=0
            const float dXI = sigxi * dw;
            const float dD  = (Zst * fabsf(Zst) / (x2 * x2)) * (kDT / 0.09f) + dXI;
            Wst  += dw;
            Dst  += dD;
            XIst += dXI;
            Zst  += dZ;
            // Xst unchanged: dX = -X_RATE*X*DT + NU*dW == 0
        }
    }
    out[path] = Zst;
}

__global__ void init_ws(float* __restrict__ ws_f, unsigned int nelem)
{
    unsigned int i = blockIdx.x * blockDim.x + threadIdx.x;
    if (i < nelem) ws_f[i] = 0.f;
}

extern "C" void kernel_launch(void* const* d_in, const int* in_sizes, int n_in,
                              void* d_out, int out_size, void* d_ws, size_t ws_size,
                              hipStream_t stream)
{
    const float* dW  = (const float*)d_in[0];
    const float* W1  = (const float*)d_in[1];
    // d_in[2] = b1: exactly cancelled by BatchNorm mean subtraction (unused)
    const float* g1  = (const float*)d_in[3];
    const float* be1 = (const float*)d_in[4];
    const float* W2  = (const float*)d_in[5];
    const float* b2  = (const float*)d_in[6];
    float* out = (float*)d_out;

    // workspace layout: gsum[501*16] | gsq[501*16] (at +8192 floats) | done[501]
    float* ws_f = (float*)d_ws;
    float* gsum = ws_f;
    float* gsq  = ws_f + 8192;
    unsigned int* done = (unsigned int*)(ws_f + 16384);

    const unsigned int zelems = 16384 + 512;  // sums + counters
    init_ws<<<(zelems + 255) / 256, 256, 0, stream>>>(ws_f, zelems);
    sde_persistent<<<NBLK, BLOCK, 0, stream>>>(dW, W1, g1, be1, W2, b2,
                                               out, gsum, gsq, done);
}